// self_attention_66743791780154
// MI455X (gfx1250) — compile-verified
//
#include <hip/hip_runtime.h>
#include <math.h>

// ---------------------------------------------------------------------------
// Problem constants (from reference)
// ---------------------------------------------------------------------------
static constexpr int kB   = 4;      // batch
static constexpr int kCin = 32;     // input channels
static constexpr int kC   = 64;     // hidden channels
static constexpr int kH   = 4;      // heads
static constexpr int kCh  = 16;     // head out channels
static constexpr int kN   = 8192;   // points
static constexpr int SPLITS = 8;    // n-splits per (b,h) in k_heads
#define EPS_VN   1e-6f
#define NS_VN    0.2f
#define INV_SCALE 0.125f            // 1/sqrt(64)

typedef float v2f __attribute__((ext_vector_type(2)));
typedef float v8f __attribute__((ext_vector_type(8)));

__device__ inline v8f vzero8() {
  v8f z;
#pragma unroll
  for (int i = 0; i < 8; ++i) z[i] = 0.f;
  return z;
}

// D(16x16 f32) = A(16x4 f32) * B(4x16 f32) + C
__device__ inline v8f wmma4(v2f a, v2f b, v8f c) {
  return __builtin_amdgcn_wmma_f32_16x16x4_f32(false, a, false, b, (short)0, c,
                                               false, false);
}

// ---------------------------------------------------------------------------
// Kernel 1: f = VN_LReLU(W_e . fts, D_e)   -> f workspace (b,C,3,N)
// 1 wave = one 16-point tile, all 64 out channels, all 3 spatial dims.
// ---------------------------------------------------------------------------
__global__ __launch_bounds__(128) void k_embed(const float* __restrict__ fts,
                                               const float* __restrict__ We,
                                               const float* __restrict__ De,
                                               float* __restrict__ fws) {
  __shared__ float sWe[kC * kCin];       // 2048 f
  __shared__ float sDe[kC * kC];         // 4096 f
  __shared__ float sStage[4][kC][16];    // per-wave restage tile

  const int tid = threadIdx.x;
  for (int i = tid; i < kC * kCin; i += 128) sWe[i] = We[i];
  for (int i = tid; i < kC * kC;   i += 128) sDe[i] = De[i];
  __syncthreads();

  const int wave = tid >> 5;
  const int lane = tid & 31;
  const int col  = lane & 15;   // tile column (point)
  const int half = lane >> 4;   // lane half
  const int b  = blockIdx.x / (kN / 64);
  const int n0 = (blockIdx.x % (kN / 64)) * 64 + wave * 16;
  const int n  = n0 + col;

  // ---- GEMM1: pre[d][m] = W_e(64x32) x fts tile -------------------------
  v8f pre[3][4];
#pragma unroll
  for (int d = 0; d < 3; ++d)
#pragma unroll
    for (int m = 0; m < 4; ++m) pre[d][m] = vzero8();

#pragma unroll
  for (int d = 0; d < 3; ++d) {
    for (int k = 0; k < kCin / 4; ++k) {
      const int c0 = 4 * k + 2 * half;
      const float* bp = fts + ((size_t)(b * kCin + c0) * 3 + d) * kN + n;
      v2f bf;
      bf.x = bp[0];
      bf.y = bp[3 * kN];  // next channel
#pragma unroll
      for (int m = 0; m < 4; ++m) {
        const int row = 16 * m + col;
        v2f af;
        af.x = sWe[row * kCin + c0];
        af.y = sWe[row * kCin + c0 + 1];
        pre[d][m] = wmma4(af, bf, pre[d][m]);
      }
    }
  }

  // ---- GEMM2: dd[d][m] = D_e(64x64) x pre[d]  (restage via LDS) ---------
  v8f dd[3][4];
#pragma unroll
  for (int d = 0; d < 3; ++d) {
#pragma unroll
    for (int m = 0; m < 4; ++m)
#pragma unroll
      for (int i = 0; i < 8; ++i)
        sStage[wave][16 * m + i + 8 * half][col] = pre[d][m][i];
    __syncthreads();
#pragma unroll
    for (int m = 0; m < 4; ++m) dd[d][m] = vzero8();
    for (int k = 0; k < kC / 4; ++k) {
      const int c0 = 4 * k + 2 * half;
      v2f bf;
      bf.x = sStage[wave][c0][col];
      bf.y = sStage[wave][c0 + 1][col];
#pragma unroll
      for (int m = 0; m < 4; ++m) {
        const int row = 16 * m + col;
        v2f af;
        af.x = sDe[row * kC + c0];
        af.y = sDe[row * kC + c0 + 1];
        dd[d][m] = wmma4(af, bf, dd[d][m]);
      }
    }
    __syncthreads();
  }

  // ---- VN leaky relu (elementwise on D fragments) + store f -------------
#pragma unroll
  for (int m = 0; m < 4; ++m)
#pragma unroll
    for (int i = 0; i < 8; ++i) {
      float fv[3] = {pre[0][m][i], pre[1][m][i], pre[2][m][i]};
      float dv[3] = {dd[0][m][i], dd[1][m][i], dd[2][m][i]};
      float dot = fv[0] * dv[0] + fv[1] * dv[1] + fv[2] * dv[2];
      float dsq = dv[0] * dv[0] + dv[1] * dv[1] + dv[2] * dv[2];
      float t = dot / (dsq + EPS_VN);
      bool posm = dot >= 0.f;
      const int o = 16 * m + i + 8 * half;
#pragma unroll
      for (int d = 0; d < 3; ++d) {
        float xr = fv[d] - t * dv[d];
        float val = NS_VN * fv[d] + (1.f - NS_VN) * (posm ? fv[d] : xr);
        fws[((size_t)(b * kC + o) * 3 + d) * kN + n] = val;
      }
    }
}

// ---------------------------------------------------------------------------
// Kernel 2: per (b, head, split): full head pipeline with online softmax.
// part[P][o*3+d][{m,s,acc}] per wave-partial, P = ((b*H+h)*SPLITS+split)*4+w
// ---------------------------------------------------------------------------
__global__ __launch_bounds__(128) void k_heads(
    const float* __restrict__ fws, const float* __restrict__ xyz,
    const float* __restrict__ Wq, const float* __restrict__ Dq,
    const float* __restrict__ Wr, const float* __restrict__ Dr,
    const float* __restrict__ Wstd, const float* __restrict__ Wfwd,
    const float* __restrict__ bfwd, float* __restrict__ part) {
  extern __shared__ float sm[];
  float* sWq   = sm;                 // 4096
  float* sDq   = sWq + 4096;         // 4096
  float* sDr   = sDq + 4096;         // 4096
  float* sWfw  = sDr + 4096;         // 4096
  float* sWr   = sWfw + 4096;        // 64
  float* sWst  = sWr + 64;           // 192
  float* sBf   = sWst + 192;         // 64
  float* sStage = sBf + 64;          // 4*64*16 = 4096
  float* sV    = sStage + 4096;      // 4*64*3*16 = 12288 (q, then v)
  float* sSt   = sV + 12288;         // 4*64*3*3  = 2304  (m,s,acc)

  const int tid   = threadIdx.x;
  const int split = blockIdx.x % SPLITS;
  const int h     = (blockIdx.x / SPLITS) % kH;
  const int b     = blockIdx.x / (SPLITS * kH);

  for (int i = tid; i < kC * kC; i += 128) {
    sWq[i]  = Wq[h * kC * kC + i];
    sDq[i]  = Dq[h * kC * kC + i];
    sDr[i]  = Dr[h * kC * kC + i];
    sWfw[i] = Wfwd[h * kC * kC + i];
  }
  for (int i = tid; i < kC; i += 128) {
    sWr[i] = Wr[h * kC + i];
    sBf[i] = bfwd[h * kC + i];
  }
  for (int i = tid; i < 3 * kC; i += 128) sWst[i] = Wstd[h * 3 * kC + i];
  for (int e = tid; e < 4 * kC * 3; e += 128) {
    sSt[e * 3 + 0] = -INFINITY;
    sSt[e * 3 + 1] = 0.f;
    sSt[e * 3 + 2] = 0.f;
  }
  __syncthreads();

  const int wave = tid >> 5;
  const int lane = tid & 31;
  const int col  = lane & 15;
  const int half = lane >> 4;
  float* stg = sStage + wave * (kC * 16);   // per-wave [64][16]

  const int TILES = kN / (SPLITS * 4 * 16);  // 16 tiles per wave
  for (int it = 0; it < TILES; ++it) {
    const int n0 = split * (kN / SPLITS) + (it * 4 + wave) * 16;
    const int n  = n0 + col;

    // ================= Q stage: q = VN_LReLU(Wq.f, Dq) ===================
    v8f qp[3][4];
#pragma unroll
    for (int d = 0; d < 3; ++d)
#pragma unroll
      for (int m = 0; m < 4; ++m) qp[d][m] = vzero8();
#pragma unroll
    for (int d = 0; d < 3; ++d)
      for (int k = 0; k < kC / 4; ++k) {
        const int c0 = 4 * k + 2 * half;
        const float* bp = fws + ((size_t)(b * kC + c0) * 3 + d) * kN + n;
        v2f bf; bf.x = bp[0]; bf.y = bp[3 * kN];
#pragma unroll
        for (int m = 0; m < 4; ++m) {
          const int row = 16 * m + col;
          v2f af; af.x = sWq[row * kC + c0]; af.y = sWq[row * kC + c0 + 1];
          qp[d][m] = wmma4(af, bf, qp[d][m]);
        }
      }
    v8f dq[3][4];
#pragma unroll
    for (int d = 0; d < 3; ++d) {
#pragma unroll
      for (int m = 0; m < 4; ++m)
#pragma unroll
        for (int i = 0; i < 8; ++i)
          stg[(16 * m + i + 8 * half) * 16 + col] = qp[d][m][i];
      __syncthreads();
#pragma unroll
      for (int m = 0; m < 4; ++m) dq[d][m] = vzero8();
      for (int k = 0; k < kC / 4; ++k) {
        const int c0 = 4 * k + 2 * half;
        v2f bf; bf.x = stg[c0 * 16 + col]; bf.y = stg[(c0 + 1) * 16 + col];
#pragma unroll
        for (int m = 0; m < 4; ++m) {
          const int row = 16 * m + col;
          v2f af; af.x = sDq[row * kC + c0]; af.y = sDq[row * kC + c0 + 1];
          dq[d][m] = wmma4(af, bf, dq[d][m]);
        }
      }
      __syncthreads();
    }
    // VN relu -> q, park q in sV (per-wave [64][3][16])
#pragma unroll
    for (int m = 0; m < 4; ++m)
#pragma unroll
      for (int i = 0; i < 8; ++i) {
        float fv[3] = {qp[0][m][i], qp[1][m][i], qp[2][m][i]};
        float dv[3] = {dq[0][m][i], dq[1][m][i], dq[2][m][i]};
        float dot = fv[0] * dv[0] + fv[1] * dv[1] + fv[2] * dv[2];
        float dsq = dv[0] * dv[0] + dv[1] * dv[1] + dv[2] * dv[2];
        float t = dot / (dsq + EPS_VN);
        bool posm = dot >= 0.f;
        const int o = 16 * m + i + 8 * half;
#pragma unroll
        for (int d = 0; d < 3; ++d) {
          float xr = fv[d] - t * dv[d];
          sV[((wave * kC + o) * 3 + d) * 16 + col] =
              NS_VN * fv[d] + (1.f - NS_VN) * (posm ? fv[d] : xr);
        }
      }

    // ================ POS stage: pos = VN_LReLU(wr*xyz, Dr) ==============
    float x3[3];
#pragma unroll
    for (int d = 0; d < 3; ++d) x3[d] = xyz[((size_t)b * kN + n) * 3 + d];
    v8f pp[3][4];
#pragma unroll
    for (int m = 0; m < 4; ++m)
#pragma unroll
      for (int i = 0; i < 8; ++i) {
        const float wr_o = sWr[16 * m + i + 8 * half];
#pragma unroll
        for (int d = 0; d < 3; ++d) pp[d][m][i] = wr_o * x3[d];
      }
    v8f dr[3][4];
#pragma unroll
    for (int d = 0; d < 3; ++d) {
#pragma unroll
      for (int m = 0; m < 4; ++m)
#pragma unroll
        for (int i = 0; i < 8; ++i)
          stg[(16 * m + i + 8 * half) * 16 + col] = pp[d][m][i];
      __syncthreads();
#pragma unroll
      for (int m = 0; m < 4; ++m) dr[d][m] = vzero8();
      for (int k = 0; k < kC / 4; ++k) {
        const int c0 = 4 * k + 2 * half;
        v2f bf; bf.x = stg[c0 * 16 + col]; bf.y = stg[(c0 + 1) * 16 + col];
#pragma unroll
        for (int m = 0; m < 4; ++m) {
          const int row = 16 * m + col;
          v2f af; af.x = sDr[row * kC + c0]; af.y = sDr[row * kC + c0 + 1];
          dr[d][m] = wmma4(af, bf, dr[d][m]);
        }
      }
      __syncthreads();
    }
    // VN relu -> pos (in pp), v = q + pos (into sV)
#pragma unroll
    for (int m = 0; m < 4; ++m)
#pragma unroll
      for (int i = 0; i < 8; ++i) {
        float fv[3] = {pp[0][m][i], pp[1][m][i], pp[2][m][i]};
        float dv[3] = {dr[0][m][i], dr[1][m][i], dr[2][m][i]};
        float dot = fv[0] * dv[0] + fv[1] * dv[1] + fv[2] * dv[2];
        float dsq = dv[0] * dv[0] + dv[1] * dv[1] + dv[2] * dv[2];
        float t = dot / (dsq + EPS_VN);
        bool posm = dot >= 0.f;
        const int o = 16 * m + i + 8 * half;
#pragma unroll
        for (int d = 0; d < 3; ++d) {
          float xr = fv[d] - t * dv[d];
          float pv = NS_VN * fv[d] + (1.f - NS_VN) * (posm ? fv[d] : xr);
          pp[d][m][i] = pv;
          const int vi = ((wave * kC + o) * 3 + d) * 16 + col;
          sV[vi] = sV[vi] + pv;  // v = q + pos
        }
      }

    // ============ z0[j][k] = sum_c Wstd[j,c]*pos[c,k]  (per point) =======
    float z0[3][3];
#pragma unroll
    for (int j = 0; j < 3; ++j)
#pragma unroll
      for (int kk = 0; kk < 3; ++kk) z0[j][kk] = 0.f;
#pragma unroll
    for (int m = 0; m < 4; ++m)
#pragma unroll
      for (int i = 0; i < 8; ++i) {
        const int o = 16 * m + i + 8 * half;
#pragma unroll
        for (int j = 0; j < 3; ++j) {
          const float w = sWst[j * kC + o];
#pragma unroll
          for (int kk = 0; kk < 3; ++kk) z0[j][kk] += w * pp[kk][m][i];
        }
      }
#pragma unroll
    for (int j = 0; j < 3; ++j)
#pragma unroll
      for (int kk = 0; kk < 3; ++kk)
        z0[j][kk] += __shfl_xor(z0[j][kk], 16, 32);  // combine lane halves

    // attn[c,k] = sum_j pos[c,j]*z0[j,k]  (in place over pp)
#pragma unroll
    for (int m = 0; m < 4; ++m)
#pragma unroll
      for (int i = 0; i < 8; ++i) {
        float a0 = pp[0][m][i] * z0[0][0] + pp[1][m][i] * z0[1][0] + pp[2][m][i] * z0[2][0];
        float a1 = pp[0][m][i] * z0[0][1] + pp[1][m][i] * z0[1][1] + pp[2][m][i] * z0[2][1];
        float a2 = pp[0][m][i] * z0[0][2] + pp[1][m][i] * z0[1][2] + pp[2][m][i] * z0[2][2];
        pp[0][m][i] = a0; pp[1][m][i] = a1; pp[2][m][i] = a2;
      }

    // ========== logits = Wfwd.attn + b ; online softmax accumulate =======
#pragma unroll
    for (int d = 0; d < 3; ++d) {
#pragma unroll
      for (int m = 0; m < 4; ++m)
#pragma unroll
        for (int i = 0; i < 8; ++i)
          stg[(16 * m + i + 8 * half) * 16 + col] = pp[d][m][i];
      __syncthreads();
      v8f lg[4];
#pragma unroll
      for (int m = 0; m < 4; ++m) lg[m] = vzero8();
      for (int k = 0; k < kC / 4; ++k) {
        const int c0 = 4 * k + 2 * half;
        v2f bf; bf.x = stg[c0 * 16 + col]; bf.y = stg[(c0 + 1) * 16 + col];
#pragma unroll
        for (int m = 0; m < 4; ++m) {
          const int row = 16 * m + col;
          v2f af; af.x = sWfw[row * kC + c0]; af.y = sWfw[row * kC + c0 + 1];
          lg[m] = wmma4(af, bf, lg[m]);
        }
      }
#pragma unroll
      for (int m = 0; m < 4; ++m)
#pragma unroll
        for (int i = 0; i < 8; ++i) {
          const int o = 16 * m + i + 8 * half;
          float lv = (lg[m][i] + sBf[o]) * INV_SCALE;
          float tm = lv;
#pragma unroll
          for (int msk = 1; msk <= 8; msk <<= 1)
            tm = fmaxf(tm, __shfl_xor(tm, msk, 32));  // max over 16 cols
          float w = __expf(lv - tm);
          float vv = sV[((wave * kC + o) * 3 + d) * 16 + col];
          float ts = w, ta = w * vv;
#pragma unroll
          for (int msk = 1; msk <= 8; msk <<= 1) {
            ts += __shfl_xor(ts, msk, 32);
            ta += __shfl_xor(ta, msk, 32);
          }
          if ((lane & 15) == 0) {  // one lane per half owns (o,d)
            float* st = &sSt[((wave * kC + o) * 3 + d) * 3];
            float mo = st[0], so = st[1], ao = st[2];
            float mn = fmaxf(mo, tm);
            float e0 = __expf(mo - mn), e1 = __expf(tm - mn);
            st[0] = mn;
            st[1] = so * e0 + ts * e1;
            st[2] = ao * e0 + ta * e1;
          }
        }
      __syncthreads();
    }
  }

  // ---- flush per-wave partial state --------------------------------------
  const int P = ((b * kH + h) * SPLITS + split) * 4 + wave;
  float* dst = part + (size_t)P * 576;
  for (int e = lane; e < kC * 3; e += 32) {
    dst[e * 3 + 0] = sSt[(wave * kC * 3 + e) * 3 + 0];
    dst[e * 3 + 1] = sSt[(wave * kC * 3 + e) * 3 + 1];
    dst[e * 3 + 2] = sSt[(wave * kC * 3 + e) * 3 + 2];
  }
}

// ---------------------------------------------------------------------------
// Kernel 3: combine partials, resi = acc/s, out = Wf . resi
// ---------------------------------------------------------------------------
__global__ __launch_bounds__(192) void k_final(const float* __restrict__ part,
                                               const float* __restrict__ Wf,
                                               float* __restrict__ out) {
  __shared__ float resi[kC][3];
  __shared__ float swf[kCh * kC];
  const int h = blockIdx.x % kH;
  const int b = blockIdx.x / kH;
  const int t = threadIdx.x;
  for (int i = t; i < kCh * kC; i += 192) swf[i] = Wf[h * kCh * kC + i];

  const int NP = SPLITS * 4;  // 32 partials per (b,h)
  {
    const int o = t / 3, d = t % 3;
    const int e = o * 3 + d;
    const float* pb = part + (size_t)(b * kH + h) * NP * 576;
    float mg = -INFINITY;
    for (int p = 0; p < NP; ++p) mg = fmaxf(mg, pb[p * 576 + e * 3]);
    float sg = 0.f, ag = 0.f;
    for (int p = 0; p < NP; ++p) {
      float mi = pb[p * 576 + e * 3 + 0];
      float si = pb[p * 576 + e * 3 + 1];
      float ai = pb[p * 576 + e * 3 + 2];
      float sc = __expf(mi - mg);
      sg += si * sc;
      ag += ai * sc;
    }
    resi[o][d] = ag / sg;
  }
  __syncthreads();
  if (t < kCh * 3) {
    const int oh = t / 3, d = t % 3;
    float s = 0.f;
    for (int c = 0; c < kC; ++c) s += swf[oh * kC + c] * resi[c][d];
    out[((size_t)b * (kH * kCh) + h * kCh + oh) * 3 + d] = s;
  }
}

// ---------------------------------------------------------------------------
extern "C" void kernel_launch(void* const* d_in, const int* in_sizes, int n_in,
                              void* d_out, int out_size, void* d_ws,
                              size_t ws_size, hipStream_t stream) {
  const float* fts  = (const float*)d_in[0];
  const float* xyz  = (const float*)d_in[1];
  const float* We   = (const float*)d_in[2];
  const float* De   = (const float*)d_in[3];
  const float* Wq   = (const float*)d_in[4];
  const float* Dq   = (const float*)d_in[5];
  const float* Wr   = (const float*)d_in[6];
  const float* Dr   = (const float*)d_in[7];
  const float* Wstd = (const float*)d_in[8];
  const float* Wfwd = (const float*)d_in[9];
  const float* bfwd = (const float*)d_in[10];
  const float* Wf   = (const float*)d_in[11];
  float* out = (float*)d_out;

  float* fws  = (float*)d_ws;                       // B*C*3*N floats
  float* part = fws + (size_t)kB * kC * 3 * kN;     // 512 * 576 floats

  k_embed<<<kB * (kN / 64), 128, 0, stream>>>(fts, We, De, fws);

  const size_t smem2 = (size_t)(4 * 4096 + 64 + 192 + 64 + 4096 + 12288 + 2304) * sizeof(float);
  k_heads<<<kB * kH * SPLITS, 128, smem2, stream>>>(fws, xyz, Wq, Dq, Wr, Dr,
                                                    Wstd, Wfwd, bfwd, part);

  k_final<<<kB * kH, 192, 0, stream>>>(part, Wf, out);
}